// SimpleCaptionDecoder_58643483459669
// MI455X (gfx1250) — compile-verified
//
#include <hip/hip_runtime.h>
#include <math.h>

// ---------------------------------------------------------------------------
// Greedy GRU caption decoder for MI455X (gfx1250, wave32).
//   B=64, EMBED=256, HID=512, VOCAB=32000, T=20 sequential steps.
// Dominant work: per-step logits GEMM 64x32000x512 in fp32 via
// V_WMMA_F32_16X16X4_F32 (keeps argmax bit-exact vs fp32 reference).
// W_out (65.5MB) stays resident in the 192MB L2 across the 20 steps.
// W_out staged into LDS with GLOBAL_LOAD_ASYNC_TO_LDS_B128 (ASYNCcnt-tracked)
// double buffering, overlapping prefetch of chunk kc+1 with WMMA on chunk kc.
// ---------------------------------------------------------------------------

typedef __attribute__((ext_vector_type(2))) float v2f;
typedef __attribute__((ext_vector_type(4))) float v4f;
typedef __attribute__((ext_vector_type(8))) float v8f;
typedef __attribute__((ext_vector_type(4))) int   v4i;

typedef __attribute__((address_space(1))) v4i* gp4;   // global src for async-to-LDS
typedef __attribute__((address_space(3))) v4i* lp4;   // LDS dst for async-to-LDS

#define VOCAB 32000
#define EMBED 256
#define HID   512
#define G3    1536          // 3*HID
#define BATCH 64
#define TLEN  20
#define SOS_ID 1
#define EOS_ID 2

// ---------------------------------------------------------------- init state
__global__ void k_init(int* __restrict__ pred, int* __restrict__ done) {
    int i = threadIdx.x;
    if (i < BATCH) { pred[i] = SOS_ID; done[i] = 0; }
}

// ------------------------------------------------------------- h0 = mean(L)
__global__ void k_h0(const float* __restrict__ feat, float* __restrict__ h) {
    int i = blockIdx.x * blockDim.x + threadIdx.x;   // 0..64*512-1
    int b = i >> 9, j = i & 511;
    const float* p = feat + (size_t)b * 49 * HID + j;
    float s = 0.f;
    #pragma unroll 7
    for (int l = 0; l < 49; ++l) s += p[l * HID];
    h[i] = s * (1.0f / 49.0f);
}

// ---------------------------------------------------- GRU gate GEMMs (WMMA)
// gi[b][n] = sum_k emb[pred[b]][k] * W_ih[n][k]   (K=256)
// gh[b][n] = sum_k h[b][k]         * W_hh[n][k]   (K=512)
// Block: 512 thr = 16 waves; tile M=64(batch) x N=64 gates; grid = 1536/64.
__global__ void __launch_bounds__(512) k_gru_gemm(
    const float* __restrict__ emb, const float* __restrict__ W_ih,
    const float* __restrict__ W_hh, const float* __restrict__ h,
    const int* __restrict__ pred,
    float* __restrict__ gi, float* __restrict__ gh) {
    int wave = threadIdx.x >> 5;
    int lane = threadIdx.x & 31;
    int mt = wave >> 2, nt = wave & 3;
    int n0 = blockIdx.x * 64 + nt * 16;
    int r16  = lane & 15;
    int koff = (lane >> 4) << 1;          // 0 or 2

    int m    = mt * 16 + r16;             // batch row this lane feeds (A frag)
    int nrow = n0 + r16;                  // weight row this lane feeds (B frag)
    const float* hA = h   + (size_t)m * HID;
    const float* xA = emb + (size_t)pred[m] * EMBED;
    const float* wI = W_ih + (size_t)nrow * EMBED;
    const float* wH = W_hh + (size_t)nrow * HID;

    v8f ai = {}; v8f ah = {};
    #pragma unroll 8
    for (int k = 0; k < EMBED; k += 4) {  // both GEMMs active
        v2f a0 = *(const v2f*)(hA + k + koff);
        v2f b0 = *(const v2f*)(wH + k + koff);
        ah = __builtin_amdgcn_wmma_f32_16x16x4_f32(false, a0, false, b0,
                                                   (short)0, ah, false, false);
        v2f a1 = *(const v2f*)(xA + k + koff);
        v2f b1 = *(const v2f*)(wI + k + koff);
        ai = __builtin_amdgcn_wmma_f32_16x16x4_f32(false, a1, false, b1,
                                                   (short)0, ai, false, false);
    }
    #pragma unroll 8
    for (int k = EMBED; k < HID; k += 4) { // gh only
        v2f a0 = *(const v2f*)(hA + k + koff);
        v2f b0 = *(const v2f*)(wH + k + koff);
        ah = __builtin_amdgcn_wmma_f32_16x16x4_f32(false, a0, false, b0,
                                                   (short)0, ah, false, false);
    }
    // C/D layout: vgpr i -> M = mt*16 + i + 8*(lane>>4), N = n0 + (lane&15)
    int mBase = mt * 16 + ((lane >> 4) << 3);
    int n = n0 + r16;
    #pragma unroll
    for (int i = 0; i < 8; ++i) {
        int mm = mBase + i;
        gi[(size_t)mm * G3 + n] = ai[i];
        gh[(size_t)mm * G3 + n] = ah[i];
    }
}

// ------------------------------------------------- elementwise GRU nonlinear
__global__ void k_gates(const float* __restrict__ gi, const float* __restrict__ gh,
                        const float* __restrict__ b_ih, const float* __restrict__ b_hh,
                        float* __restrict__ h) {
    int i = blockIdx.x * blockDim.x + threadIdx.x;   // 0..64*512-1
    int b = i >> 9, j = i & 511;
    const float* gib = gi + (size_t)b * G3;
    const float* ghb = gh + (size_t)b * G3;
    float ir = gib[j]           + b_ih[j];
    float hr = ghb[j]           + b_hh[j];
    float iz = gib[j + HID]     + b_ih[j + HID];
    float hz = ghb[j + HID]     + b_hh[j + HID];
    float in_ = gib[j + 2*HID]  + b_ih[j + 2*HID];
    float hn  = ghb[j + 2*HID]  + b_hh[j + 2*HID];
    float r = 1.f / (1.f + expf(-(ir + hr)));
    float z = 1.f / (1.f + expf(-(iz + hz)));
    float n = tanhf(in_ + r * hn);
    float ho = h[i];
    h[i] = (1.f - z) * n + z * ho;
}

// ------------------------------------------------------ logits GEMM (WMMA)
// logits[b][v] = h[b]·W_out[v] + b_out[v].  M=64, N=32000, K=512.
// Block: 512 thr = 16 waves, tile 64x64.
// W_out K-chunks (64 wide) double-buffered in LDS via async-to-LDS B128
// (ASYNCcnt), prefetching chunk kc+1 while WMMAs consume chunk kc.
// h fragments come straight from global (128KB total, L0/L2-hot; round-1 asm
// shows these pipeline cleanly against WMMA).
// Row pitch 68 floats: 16B-aligned B128 writes, conflict-free b64 frag reads
// (banks 4*row + {0,1}/{2,3} cover all 64 banks across the wave).
__global__ void __launch_bounds__(512) k_logits(
    const float* __restrict__ h, const float* __restrict__ Wout,
    const float* __restrict__ bout, float* __restrict__ out, int t) {
    __shared__ __align__(16) float lw[2][BATCH][68];
    int tid  = threadIdx.x;
    int wave = tid >> 5, lane = tid & 31;
    int mt = wave >> 2, nt = wave & 3;
    int v0 = blockIdx.x * 64;
    int r16 = lane & 15, koff = (lane >> 4) << 1;

    // per-thread staging assignment: 2 x 16B per chunk
    int f0row = tid >> 4;                 // rows 0..31   (rep 0)
    int f1row = f0row + 32;               // rows 32..63  (rep 1)
    int fc4   = (tid & 15) << 2;          // float4 column

    const float* hA = h + (size_t)(mt * 16 + r16) * HID;  // A-frag row (global)

    // prefetch chunk 0
    __builtin_amdgcn_global_load_async_to_lds_b128(
        (gp4)(Wout + (size_t)(v0 + f0row) * HID + fc4),
        (lp4)&lw[0][f0row][fc4], 0, 0);
    __builtin_amdgcn_global_load_async_to_lds_b128(
        (gp4)(Wout + (size_t)(v0 + f1row) * HID + fc4),
        (lp4)&lw[0][f1row][fc4], 0, 0);
    __builtin_amdgcn_s_wait_asynccnt(0);
    __syncthreads();

    v8f acc = {};
    for (int kc = 0; kc < 8; ++kc) {
        int cur = kc & 1;
        if (kc < 7) {                     // prefetch next chunk into other buf
            int k1 = (kc + 1) * 64;
            __builtin_amdgcn_global_load_async_to_lds_b128(
                (gp4)(Wout + (size_t)(v0 + f0row) * HID + k1 + fc4),
                (lp4)&lw[cur ^ 1][f0row][fc4], 0, 0);
            __builtin_amdgcn_global_load_async_to_lds_b128(
                (gp4)(Wout + (size_t)(v0 + f1row) * HID + k1 + fc4),
                (lp4)&lw[cur ^ 1][f1row][fc4], 0, 0);
        }
        int kbase = kc * 64;
        #pragma unroll
        for (int kk = 0; kk < 64; kk += 4) {
            v2f a = *(const v2f*)(hA + kbase + kk + koff);
            v2f b = *(const v2f*)&lw[cur][nt * 16 + r16][kk + koff];
            acc = __builtin_amdgcn_wmma_f32_16x16x4_f32(false, a, false, b,
                                                        (short)0, acc, false, false);
        }
        __builtin_amdgcn_s_wait_asynccnt(0);
        __syncthreads();
    }

    int v  = v0 + nt * 16 + r16;
    float bb = bout[v];
    int mBase = mt * 16 + ((lane >> 4) << 3);
    float* lg = out + (size_t)BATCH * TLEN;          // logits base after toks
    #pragma unroll
    for (int i = 0; i < 8; ++i) {
        int m = mBase + i;                           // batch row
        lg[((size_t)(m * TLEN + t)) * VOCAB + v] = acc[i] + bb;
    }
}

// ------------------------------------------------------- argmax + EOS logic
__global__ void k_argmax(const float* __restrict__ lg_base, float* __restrict__ toks,
                         int* __restrict__ pred, int* __restrict__ done, int t) {
    __shared__ float sv[256];
    __shared__ int   si[256];
    int b = blockIdx.x, tid = threadIdx.x;
    const float* lg = lg_base + ((size_t)(b * TLEN + t)) * VOCAB;
    float best = -__builtin_inff();
    int   bi   = 0x7fffffff;
    for (int v = tid; v < VOCAB; v += 256) {
        float val = lg[v];
        if (val > best) { best = val; bi = v; }      // strict >: first index wins
    }
    sv[tid] = best; si[tid] = bi;
    __syncthreads();
    for (int s = 128; s > 0; s >>= 1) {
        if (tid < s) {
            float ov = sv[tid + s]; int oi = si[tid + s];
            if (ov > sv[tid] || (ov == sv[tid] && oi < si[tid])) {
                sv[tid] = ov; si[tid] = oi;
            }
        }
        __syncthreads();
    }
    if (tid == 0) {
        int p = si[0];
        int d = done[b];
        int is_eos = (p == EOS_ID);
        toks[b * TLEN + t] = (float)((d || is_eos) ? 0 : p);
        done[b] = d | is_eos;
        pred[b] = p;
    }
}

// ---------------------------------------------------------------------------
extern "C" void kernel_launch(void* const* d_in, const int* in_sizes, int n_in,
                              void* d_out, int out_size, void* d_ws, size_t ws_size,
                              hipStream_t stream) {
    const float* feat  = (const float*)d_in[0];  // [64,49,512]
    const float* emb   = (const float*)d_in[1];  // [32000,256]
    const float* W_ih  = (const float*)d_in[2];  // [1536,256]
    const float* W_hh  = (const float*)d_in[3];  // [1536,512]
    const float* b_ih  = (const float*)d_in[4];  // [1536]
    const float* b_hh  = (const float*)d_in[5];  // [1536]
    const float* W_out = (const float*)d_in[6];  // [32000,512]
    const float* b_out = (const float*)d_in[7];  // [32000]
    // d_in[8] = max_len (==20, hardcoded: not host-readable under graph capture)

    float* out = (float*)d_out;                  // [64*20] toks, then [64,20,32000]
    float* ws  = (float*)d_ws;
    float* h   = ws;                             // 64*512
    float* gi  = ws + 32768;                     // 64*1536
    float* gh  = ws + 131072;                    // 64*1536
    int*   pred = (int*)(ws + 229376);           // 64
    int*   done = (int*)(ws + 229440);           // 64

    k_init<<<1, 64, 0, stream>>>(pred, done);
    k_h0<<<128, 256, 0, stream>>>(feat, h);

    for (int t = 0; t < TLEN; ++t) {
        k_gru_gemm<<<G3 / 64, 512, 0, stream>>>(emb, W_ih, W_hh, h, pred, gi, gh);
        k_gates<<<128, 256, 0, stream>>>(gi, gh, b_ih, b_hh, h);
        k_logits<<<VOCAB / 64, 512, 0, stream>>>(h, W_out, b_out, out, t);
        k_argmax<<<BATCH, 256, 0, stream>>>(out + (size_t)BATCH * TLEN, out, pred, done, t);
    }
}